// SeqDistanceSelfAttention_75136157876438
// MI455X (gfx1250) — compile-verified
//
#include <hip/hip_runtime.h>
#include <math.h>

typedef __attribute__((ext_vector_type(2))) float v2f;
typedef __attribute__((ext_vector_type(4))) float v4f;
typedef __attribute__((ext_vector_type(8))) float v8f;

constexpr int SEQ   = 15;
constexpr int DIM   = 512;
constexpr int UNITS = 32;

constexpr int XS_STRIDE = 516;  // 516*4 bytes: 16B aligned rows, bank-friendly (4*m pattern)
constexpr int QK_STRIDE = 66;
constexpr int AM_STRIDE = 18;   // even -> 8B aligned v2f A-fragments, conflict-free

__global__ __launch_bounds__(128)
void seqdist_attn_fused(const float* __restrict__ x,
                        const float* __restrict__ Wt,
                        const float* __restrict__ Wx,
                        const float* __restrict__ bh,
                        const float* __restrict__ Wa,
                        const float* __restrict__ ba,
                        float* __restrict__ out)
{
    __shared__ float xs[16 * XS_STRIDE];   // x[b] padded to 16 rows
    __shared__ float qk[16 * QK_STRIDE];   // cols 0..31 = q, 32..63 = k
    __shared__ float am[16 * AM_STRIDE];   // emission / attention matrix (16x16 padded)
    __shared__ float wbuf[65];             // [0..31]=Wa, [32..63]=bh, [64]=ba

    const int tid  = threadIdx.x;
    const int lane = tid & 31;
    const int wave = tid >> 5;
    const int m    = lane & 15;   // A row / B-D column index within tile
    const int h    = lane >> 4;   // lane half -> K pair / C row-half
    const long long xbase = (long long)blockIdx.x * (SEQ * DIM);

    // ---------------- Phase 0: stage x[b] into LDS, zero pads, cache small vecs
    for (int i = tid; i < (SEQ * DIM) / 4; i += 128) {
        int j = i * 4;
        int r = j >> 9;          // j / DIM
        int c = j & (DIM - 1);   // j % DIM
        v4f v = *(const v4f*)(x + xbase + j);
        *(v4f*)(&xs[r * XS_STRIDE + c]) = v;
    }
    for (int c = tid; c < DIM; c += 128) xs[15 * XS_STRIDE + c] = 0.0f;   // pad row
    for (int i = tid; i < 16 * AM_STRIDE; i += 128) am[i] = 0.0f;        // pad A matrix
    if (tid < 32)       wbuf[tid] = Wa[tid];
    else if (tid < 64)  wbuf[tid] = bh[tid - 32];
    else if (tid == 64) wbuf[64]  = ba[0];
    __syncthreads();

    // ---------------- Phase 1: qk[16 x 64] = xs[16 x 512] @ [Wt | Wx]  (WMMA f32 16x16x4)
    {
        const float* Wp  = (wave < 2) ? Wt : Wx;     // wave-uniform
        const int    wcb = (wave & 1) * 16;          // column base inside Wt/Wx
        v8f acc = {};
        for (int k0 = 0; k0 < DIM; k0 += 4) {
            const int ka = k0 + 2 * h;
            v2f a = *(const v2f*)(&xs[m * XS_STRIDE + ka]);  // ds_load_b64, 8B aligned
            v2f b;
            b.x = Wp[ka * UNITS + wcb + m];
            b.y = Wp[(ka + 1) * UNITS + wcb + m];
            acc = __builtin_amdgcn_wmma_f32_16x16x4_f32(false, a, false, b,
                                                        (short)0, acc, false, false);
        }
        #pragma unroll
        for (int r = 0; r < 8; ++r) {
            int row = h * 8 + r;
            if (row < SEQ) qk[row * QK_STRIDE + wave * 16 + m] = acc[r];
        }
    }
    __syncthreads();

    // ---------------- Phase 2: emission e[s,t] = Wa . tanh(q_s + k_t + bh) + ba + mask
    for (int p = tid; p < SEQ * SEQ; p += 128) {
        int s = p / SEQ, t = p % SEQ;
        float acc = wbuf[64];
        #pragma unroll
        for (int u = 0; u < UNITS; ++u) {
            float hv = qk[s * QK_STRIDE + u] + qk[t * QK_STRIDE + 32 + u] + wbuf[32 + u];
            acc += wbuf[u] * tanhf(hv);
        }
        int d = s - t; if (d < 0) d = -d;
        acc += (d == 0) ? -10000.0f : -(float)d;
        am[s * AM_STRIDE + t] = acc;
    }
    __syncthreads();

    // softmax over keys, one row per thread (rows 0..14)
    if (tid < SEQ) {
        float mx = -1e30f;
        #pragma unroll
        for (int t = 0; t < SEQ; ++t) mx = fmaxf(mx, am[tid * AM_STRIDE + t]);
        float ex[SEQ];
        float sum = 0.0f;
        #pragma unroll
        for (int t = 0; t < SEQ; ++t) {
            ex[t] = __expf(am[tid * AM_STRIDE + t] - mx);
            sum += ex[t];
        }
        float inv = 1.0f / sum;
        #pragma unroll
        for (int t = 0; t < SEQ; ++t) am[tid * AM_STRIDE + t] = ex[t] * inv;
    }
    __syncthreads();

    // ---------------- Phase 3: v[16 x 512] = am[16 x 16] @ xs[16 x 512]  (WMMA f32 16x16x4)
    for (int ct = wave * 8; ct < wave * 8 + 8; ++ct) {
        const int cb = ct * 16;
        v8f acc = {};
        #pragma unroll
        for (int k0 = 0; k0 < 16; k0 += 4) {
            const int ka = k0 + 2 * h;
            v2f a = *(const v2f*)(&am[m * AM_STRIDE + ka]);  // 8B aligned (ka even)
            v2f b;
            b.x = xs[ka * XS_STRIDE + cb + m];
            b.y = xs[(ka + 1) * XS_STRIDE + cb + m];
            acc = __builtin_amdgcn_wmma_f32_16x16x4_f32(false, a, false, b,
                                                        (short)0, acc, false, false);
        }
        #pragma unroll
        for (int r = 0; r < 8; ++r) {
            int row = h * 8 + r;
            if (row < SEQ) out[xbase + (long long)row * DIM + cb + m] = acc[r];
        }
    }
}

extern "C" void kernel_launch(void* const* d_in, const int* in_sizes, int n_in,
                              void* d_out, int out_size, void* d_ws, size_t ws_size,
                              hipStream_t stream) {
    (void)in_sizes; (void)n_in; (void)out_size; (void)d_ws; (void)ws_size;
    const float* x  = (const float*)d_in[0];
    const float* Wt = (const float*)d_in[1];
    const float* Wx = (const float*)d_in[2];
    const float* bh = (const float*)d_in[3];
    const float* Wa = (const float*)d_in[4];
    const float* ba = (const float*)d_in[5];
    float* out = (float*)d_out;

    dim3 grid(16384);
    dim3 block(128);
    hipLaunchKernelGGL(seqdist_attn_fused, grid, block, 0, stream,
                       x, Wt, Wx, bh, Wa, ba, out);
}